// SingleChannelInterp_62294205662084
// MI455X (gfx1250) — compile-verified
//
#include <hip/hip_runtime.h>

typedef __attribute__((ext_vector_type(2))) float v2f;
typedef __attribute__((ext_vector_type(8))) float v8f;

#define T_N 256
#define R_N 192
#define REF_STEP (48.0f / 191.0f)

// One workgroup per (b, d) pair. 256 threads = 8 wave32.
// Phase 1: stage x, log(m), ts rows into LDS.
// Phase 2: per-r row maxima for both temperature paths (numerical stability).
// Phase 3: WMMA f32 16x16x4 reduction over t: D = E(16r x 4t) * [x | 1 | 0...](4t x 16)
//          accumulated over 64 K-chunks -> col0 = weighted sum, col1 = denominator.
__global__ __launch_bounds__(256) void interp_wmma_kernel(
    const float* __restrict__ x, const float* __restrict__ kernel,
    float* __restrict__ out) {
  __shared__ float sh_x[T_N];
  __shared__ float sh_lm[T_N];
  __shared__ float sh_ts[T_N];
  __shared__ float sh_mx[2][R_N];
  __shared__ float sh_red[8][2][16];  // per-wave [ysum|den][row]

  const int tid = threadIdx.x;
  const int bd  = blockIdx.x;      // 0..1023
  const int b   = bd >> 4;
  const int d   = bd & 15;

  // alpha = softplus(kernel[d]), numerically stable
  const float k = kernel[d];
  const float alpha = (k > 0.0f) ? (k + log1pf(__expf(-k))) : log1pf(__expf(k));

  const float* base = x + (size_t)b * 64 * T_N;
  const float* px = base + (size_t)d * T_N;          // values
  const float* pm = base + (size_t)(16 + d) * T_N;   // mask
  const float* pt = base + (size_t)(32 + d) * T_N;   // timestamps

  if (tid == 0) {  // gfx1250 global_prefetch_b8 path
    __builtin_prefetch(px, 0, 1);
    __builtin_prefetch(pm, 0, 1);
    __builtin_prefetch(pt, 0, 1);
  }

  // ---- Phase 1: stage into LDS ----
  sh_x[tid]  = px[tid];
  sh_lm[tid] = __logf(fmaxf(pm[tid], 1e-30f));
  sh_ts[tid] = pt[tid];
  __syncthreads();

  // ---- Phase 2: per-r maxima for s (path 0) and s10 (path 1) ----
  if (tid < R_N) {
    const float refr = tid * REF_STEP;
    float mx1 = -3.0e38f, mx10 = -3.0e38f;
    for (int t = 0; t < T_N; ++t) {
      const float dz = sh_ts[t] - refr;
      const float z  = alpha * dz * dz;
      const float lm = sh_lm[t];
      mx1  = fmaxf(mx1,  lm - z);
      mx10 = fmaxf(mx10, lm - 10.0f * z);
    }
    sh_mx[0][tid] = mx1;
    sh_mx[1][tid] = mx10;
  }
  __syncthreads();

  // ---- Phase 3: WMMA reductions ----
  const int wave  = tid >> 5;
  const int lane  = tid & 31;
  const int m16   = lane & 15;   // row (A/D) / column (B/D) index
  const int khalf = lane >> 4;   // selects K pair {0,1} vs {2,3}

  // Branchless B-column selectors (hoisted out of the hot loop so the
  // inner loop has no divergent exec-masked LDS loads):
  //   col0 -> x[t], col1 -> 1.0, cols 2..15 -> 0.
  const float selx = (m16 == 0) ? 1.0f : 0.0f;
  const float sel1 = (m16 == 1) ? 1.0f : 0.0f;

  for (int i = 0; i < 3; ++i) {
    const int job = wave + 8 * i;   // 0..23
    const int p   = job / 12;       // temperature path
    const int R   = job % 12;       // r tile
    const int r   = R * 16 + m16;
    const float refr  = r * REF_STEP;
    const float scale = (p == 0) ? alpha : 10.0f * alpha;
    const float mx    = sh_mx[p][r];

    v8f acc = {};
    for (int chunk = 0; chunk < 64; ++chunk) {
      const int t0 = chunk * 4 + khalf * 2;
      const float d0 = sh_ts[t0]     - refr;
      const float d1 = sh_ts[t0 + 1] - refr;
      const float x0 = sh_x[t0];
      const float x1 = sh_x[t0 + 1];
      v2f av, bv;
      // A[16r x 4t]: exp weights, computed on the fly in f32
      av.x = __expf(sh_lm[t0]     - scale * d0 * d0 - mx);
      av.y = __expf(sh_lm[t0 + 1] - scale * d1 * d1 - mx);
      // B[4t x 16n]: col0 = x[t], col1 = 1, others 0 (branchless)
      bv.x = fmaf(selx, x0, sel1);
      bv.y = fmaf(selx, x1, sel1);
      acc = __builtin_amdgcn_wmma_f32_16x16x4_f32(
          /*neg_a=*/false, av, /*neg_b=*/false, bv,
          /*c_mod=*/(short)0, acc, /*reuse_a=*/false, /*reuse_b=*/false);
    }

    // Extract D columns 0 (ysum) and 1 (den); D layout: lane n = L&15,
    // VGPR j holds row j + 8*(L>=16).
    if (m16 < 2) {
#pragma unroll
      for (int j = 0; j < 8; ++j)
        sh_red[wave][m16][j + 8 * khalf] = acc[j];
    }
    __syncthreads();

    if (lane < 16) {
      const int r2   = R * 16 + lane;
      const float dn = sh_red[wave][1][lane];
      const float ys = sh_red[wave][0][lane];
      const size_t ob = (size_t)b * 48 * R_N;
      const int ch_y = (p == 0) ? d : (32 + d);
      out[ob + (size_t)ch_y * R_N + r2] = ys / dn;
      if (p == 0)  // logsumexp output channel
        out[ob + (size_t)(16 + d) * R_N + r2] = sh_mx[0][r2] + __logf(dn);
    }
    __syncthreads();
  }
}

extern "C" void kernel_launch(void* const* d_in, const int* in_sizes, int n_in,
                              void* d_out, int out_size, void* d_ws, size_t ws_size,
                              hipStream_t stream) {
  const float* x      = (const float*)d_in[0];  // (64, 64, 256) f32
  const float* kernel = (const float*)d_in[1];  // (16,) f32
  float* out          = (float*)d_out;          // (64, 48, 192) f32
  (void)in_sizes; (void)n_in; (void)out_size; (void)d_ws; (void)ws_size;

  interp_wmma_kernel<<<dim3(64 * 16), dim3(256), 0, stream>>>(x, kernel, out);
}